// distance_6133213299127
// MI455X (gfx1250) — compile-verified
//
#include <hip/hip_runtime.h>
#include <stdint.h>

#define NROW 2048
#define NDIM 128

typedef __attribute__((ext_vector_type(2))) float v2f;
typedef __attribute__((ext_vector_type(8))) float v8f;

// Truncating a flat (generic) pointer to a __shared__ object yields the raw
// LDS byte address (the low 32 bits of the LDS-aperture address).
__device__ __forceinline__ unsigned lds_addr_of(const void* p) {
    return (unsigned)(uintptr_t)p;
}

__global__ __launch_bounds__(256, 2) void sparse_l1_exp_rownorm(
    const float* __restrict__ feats,
    const float* __restrict__ adj,
    float* __restrict__ out)
{
    __shared__ __align__(16) float s_adj[NROW];     // staged adjacency row (async)
    __shared__ __align__(16) float s_e[NROW];       // unnormalized e row
    __shared__ __align__(16) float s_fi[NDIM];      // feats[row]
    __shared__ int   s_list[NROW];                  // compacted nonzero columns
    __shared__ int   s_cnt;
    __shared__ float s_partial[8];

    const int tid  = threadIdx.x;
    const int wave = tid >> 5;
    const int lane = tid & 31;
    const int row  = blockIdx.x;

    // ---- 1) async-stage adj row into LDS (GLOBAL_LOAD_ASYNC_TO_LDS, ASYNCcnt) ----
    {
        const unsigned long long gbase = (unsigned long long)(adj + (size_t)row * NROW);
        const unsigned lds0 = lds_addr_of(&s_adj[0]);
        #pragma unroll
        for (int k = 0; k < 2; ++k) {
            unsigned byteoff = (unsigned)(tid * 16 + k * 4096);   // 256 thr * 2 * 16B = 8KB
            unsigned ldst    = lds0 + byteoff;
            asm volatile("global_load_async_to_lds_b128 %0, %1, %2"
                         :: "v"(ldst), "v"(byteoff), "s"(gbase)
                         : "memory");
        }
    }

    if (tid == 0) s_cnt = 0;
    // load feats[row] into LDS while the async copy is in flight
    if (tid < NDIM / 4) {
        ((float4*)s_fi)[tid] = ((const float4*)(feats + (size_t)row * NDIM))[tid];
    }

    asm volatile("s_wait_asynccnt 0x0" ::: "memory");
    __syncthreads();

    // ---- 2) compact nonzero columns; zero s_e ----
    #pragma unroll
    for (int k = 0; k < 8; ++k) {
        int j = tid + k * 256;
        float a = s_adj[j];
        s_e[j] = 0.0f;
        if (a != 0.0f) {
            int p = atomicAdd(&s_cnt, 1);
            s_list[p] = j;
        }
    }
    __syncthreads();
    const int count = s_cnt;

    // ---- 3) one wave per nonzero pair: 128-dim L1 distance + exp kernel ----
    const float4 cv = ((const float4*)s_fi)[lane];
    for (int idx = wave; idx < count; idx += 8) {
        const int j = s_list[idx];
        const float4 fv = ((const float4*)(feats + (size_t)j * NDIM))[lane];
        float d = fabsf(fv.x - cv.x) + fabsf(fv.y - cv.y) +
                  fabsf(fv.z - cv.z) + fabsf(fv.w - cv.w);
        #pragma unroll
        for (int off = 16; off > 0; off >>= 1)
            d += __shfl_xor(d, off, 32);
        if (lane == 0) {
            s_e[j] = s_adj[j] * __expf(-0.01f * d);
        }
    }
    __syncthreads();

    // ---- 4) row sum via WMMA (V_WMMA_F32_16X16X4_F32 cross-lane reduction) ----
    float v[8];
    float part = 0.0f;
    #pragma unroll
    for (int k = 0; k < 8; ++k) {
        v[k] = s_e[tid + k * 256];
        part += v[k];
    }
    // A (16x4): vgpr0 = per-lane partial -> A[m,0]=p[m], A[m,2]=p[m+16]; vgpr1 = 0.
    // B (4x16) = ones. D[m,n] = p[m] + p[m+16] for every n.
    v2f a; a[0] = part; a[1] = 0.0f;
    v2f b; b[0] = 1.0f; b[1] = 1.0f;
    v8f c = {};
    v8f dacc = __builtin_amdgcn_wmma_f32_16x16x4_f32(
        /*neg_a=*/false, a, /*neg_b=*/false, b,
        /*c_mod=*/(short)0, c, /*reuse_a=*/false, /*reuse_b=*/false);
    float s = ((dacc[0] + dacc[1]) + (dacc[2] + dacc[3])) +
              ((dacc[4] + dacc[5]) + (dacc[6] + dacc[7]));
    // lanes 0-15 hold sum of p[0..7]+p[16..23]; lanes 16-31 the complement half
    float wave_total = s + __shfl_xor(s, 16, 32);
    if (lane == 0) s_partial[wave] = wave_total;
    __syncthreads();

    float denom = ((s_partial[0] + s_partial[1]) + (s_partial[2] + s_partial[3])) +
                  ((s_partial[4] + s_partial[5]) + (s_partial[6] + s_partial[7]));
    denom = fmaxf(denom, 1e-12f);
    const float inv = 1.0f / denom;

    // ---- 5) normalized, fully-dense coalesced row write ----
    float* orow = out + (size_t)row * NROW;
    #pragma unroll
    for (int k = 0; k < 8; ++k) {
        orow[tid + k * 256] = v[k] * inv;
    }
}

extern "C" void kernel_launch(void* const* d_in, const int* in_sizes, int n_in,
                              void* d_out, int out_size, void* d_ws, size_t ws_size,
                              hipStream_t stream) {
    (void)in_sizes; (void)n_in; (void)out_size; (void)d_ws; (void)ws_size;
    const float* feats = (const float*)d_in[0];
    const float* adj   = (const float*)d_in[1];
    float* out         = (float*)d_out;
    sparse_l1_exp_rownorm<<<NROW, 256, 0, stream>>>(feats, adj, out);
}